// ExpectedKLDivergence_884763263269
// MI455X (gfx1250) — compile-verified
//
#include <hip/hip_runtime.h>
#include <stdint.h>

// Problem constants (must match reference)
#define B_ROWS   512
#define S_LEN    32768

// Tiling
#define THREADS     256
#define TILE        4096                 // posterior pairs per block (32 KB LDS)
#define PER_THREAD  (TILE / THREADS)     // 16 positions per thread
#define UNITS_PT    (PER_THREAD / 2)     // 8 async b128 units per thread
#define CHUNKS      (S_LEN / TILE)       // 8 chunks per row
#define NBLOCKS     (B_ROWS * CHUNKS)    // 4096 blocks -> 4096 fp32 partials in d_ws

// log2 constants: alpha=0.1, beta=0.9
#define L2_01  (-3.3219280948873623f)    // log2(0.1) = log2(alpha) = log2(1-beta)
#define L2_09  (-0.15200309344504997f)   // log2(0.9) = log2(1-alpha) = log2(beta)
#define LN2_F  (0.6931471805599453f)

__device__ __forceinline__ float fast_log2(float x) {
    return __builtin_amdgcn_logf(x);     // v_log_f32
}

// Async global->LDS copy of 16 bytes (CDNA5 data mover path, ASYNCcnt-tracked).
__device__ __forceinline__ void async_copy_b128(uint32_t lds_byte_addr, const void* gaddr) {
    asm volatile("global_load_async_to_lds_b128 %0, %1, off"
                 :: "v"(lds_byte_addr), "v"(gaddr)
                 : "memory");
}

__device__ __forceinline__ void wait_asynccnt0() {
    asm volatile("s_wait_asynccnt 0" ::: "memory");
}

__global__ __launch_bounds__(THREADS)
void ekl_main_kernel(const float* __restrict__ post,
                     const long long* __restrict__ length,
                     float* __restrict__ partials)
{
    __shared__ float2 tile[TILE];     // pairs [start .. start+lenLoc)
    __shared__ float2 prevPair;       // pair at start-1 (chunk > 0)
    __shared__ float  red[THREADS];

    const int bid   = blockIdx.x;
    const int row   = bid / CHUNKS;
    const int chunk = bid % CHUNKS;
    const int tid   = threadIdx.x;
    const int start = chunk * TILE;

    const long long len = length[row];

    // Entire chunk masked out: contribute exactly 0, skip all memory traffic.
    // Uniform per block; deterministic given inputs.
    if ((long long)start >= len) {
        if (tid == 0) partials[bid] = 0.0f;
        return;
    }

    // Pairs actually needed in this tile (1..TILE), 32-bit from here on.
    const long long rem = len - (long long)start;
    const int lenLoc = (rem > (long long)TILE) ? TILE : (int)rem;
    const int units  = (lenLoc + 1) >> 1;            // ceil(lenLoc/2) b128 units
    const int jmin   = (chunk == 0) ? 1 : 0;         // s >= 1 condition

    const float* rowBase = post + (((size_t)row * S_LEN + start) * 2);

    // ---- Stage only the live prefix into LDS via async B128 loads ----
    const uint32_t ldsBase = (uint32_t)(uintptr_t)(void*)tile;
    #pragma unroll
    for (int j = 0; j < UNITS_PT; ++j) {
        const int u = tid + j * THREADS;             // b128 unit index
        if (u < units) {
            async_copy_b128(ldsBase + (uint32_t)u * 16u,
                            (const void*)(rowBase + (size_t)u * 4));
        }
    }
    if (tid == 0) {
        prevPair = (chunk > 0)
                 ? ((const float2*)post)[(size_t)row * S_LEN + start - 1]
                 : make_float2(1.0f, 1.0f);          // unused when chunk==0 (jmin=1)
    }
    wait_asynccnt0();
    __syncthreads();

    // ---- Compute: interleaved mapping keeps LDS reads conflict-free ----
    float acc = 0.0f;
    #pragma unroll
    for (int k = 0; k < PER_THREAD; ++k) {
        const int j = tid + k * THREADS;             // slot in tile
        if (j < lenLoc) {
            const float2 c = tile[j];
            const float2 p = (j > 0) ? tile[j - 1] : prevPair;

            const float lc0 = fast_log2(c.x);
            const float lc1 = fast_log2(c.y);
            const float A2  = c.y * lc1 + c.x * lc0;     // c1*lg c1 + c0*lg c0
            const float X   = p.y * c.y + p.x * c.x;     // coeff of lg(beta)
            const float Y   = p.y * c.x + p.x * c.y;     // coeff of lg(1-beta)
            const float div = (p.x + p.y) * A2 - L2_09 * X - L2_01 * Y;

            if (j >= jmin) acc += div;                   // branch mask: no NaN leak
        }
    }

    // First-position term (unmasked), once per row.
    if (tid == 0 && chunk == 0) {
        const float2 f = tile[0];
        const float lp0 = fast_log2(f.x);
        const float lp1 = fast_log2(f.y);
        acc += (f.y * lp1 + f.x * lp0) - f.y * L2_01 - f.x * L2_09;
    }

    // ---- Deterministic fixed-tree block reduction ----
    red[tid] = acc;
    __syncthreads();
    #pragma unroll
    for (int off = THREADS / 2; off > 0; off >>= 1) {
        if (tid < off) red[tid] += red[tid + off];
        __syncthreads();
    }
    if (tid == 0) partials[bid] = red[0];
}

__global__ __launch_bounds__(THREADS)
void ekl_reduce_kernel(const float* __restrict__ partials, float* __restrict__ out)
{
    __shared__ float red[THREADS];
    float a = 0.0f;
    #pragma unroll
    for (int i = 0; i < NBLOCKS / THREADS; ++i)
        a += partials[threadIdx.x + i * THREADS];
    red[threadIdx.x] = a;
    __syncthreads();
    #pragma unroll
    for (int off = THREADS / 2; off > 0; off >>= 1) {
        if (threadIdx.x < off) red[threadIdx.x] += red[threadIdx.x + off];
        __syncthreads();
    }
    if (threadIdx.x == 0) out[0] = red[0] * (LN2_F / (float)B_ROWS);
}

extern "C" void kernel_launch(void* const* d_in, const int* in_sizes, int n_in,
                              void* d_out, int out_size, void* d_ws, size_t ws_size,
                              hipStream_t stream) {
    (void)in_sizes; (void)n_in; (void)out_size; (void)ws_size;
    const float*     post = (const float*)d_in[0];       // (512, 32768, 2) fp32
    const long long* len  = (const long long*)d_in[1];   // (512,) int64
    float* partials = (float*)d_ws;                      // 4096 floats (16 KB)

    ekl_main_kernel<<<NBLOCKS, THREADS, 0, stream>>>(post, len, partials);
    ekl_reduce_kernel<<<1, THREADS, 0, stream>>>(partials, (float*)d_out);
}